// BlockQuantizer_12919261626615
// MI455X (gfx1250) — compile-verified
//
#include <hip/hip_runtime.h>
#include <math.h>

typedef float v2f __attribute__((ext_vector_type(2)));
typedef float v4f __attribute__((ext_vector_type(4)));
typedef float v8f __attribute__((ext_vector_type(8)));

#define BROWS   32
#define CHUNKS  128     // partial-reduction blocks per row
#define TPB     256     // 8 waves (wave32)

// DANUQ 4-bit codebook (15 entries, padded to 16)
__constant__ float Q4c[16] = {
    -2.6536f, -1.9735f, -1.508f, -1.149f, -0.8337f, -0.5439f, -0.2686f, 0.0f,
     0.2686f,  0.5439f,  0.8337f, 1.149f,  1.508f,   1.9735f,  2.6536f, 0.0f};

// ---------------------------------------------------------------------------
// Pass 1: per-(row,chunk) partials: {sum, sumsq, min, max}
// ---------------------------------------------------------------------------
__global__ __launch_bounds__(TPB)
void stats_kernel(const float* __restrict__ x, float* __restrict__ partials, int n4) {
    const int row   = blockIdx.y;
    const int chunk = blockIdx.x;
    const int chunk4 = (n4 + CHUNKS - 1) / CHUNKS;
    const v4f* __restrict__ xr = (const v4f*)x + (long long)row * n4;

    int lo = chunk * chunk4;
    int hi = lo + chunk4; if (hi > n4) hi = n4;

    float s = 0.f, ss = 0.f;
    float mn = __builtin_inff(), mx = -__builtin_inff();

    for (int i = lo + (int)threadIdx.x; i < hi; i += TPB) {
        v4f v = xr[i];
        s += (v.x + v.y) + (v.z + v.w);
        ss = fmaf(v.x, v.x, ss); ss = fmaf(v.y, v.y, ss);
        ss = fmaf(v.z, v.z, ss); ss = fmaf(v.w, v.w, ss);
        mn = fminf(mn, fminf(fminf(v.x, v.y), fminf(v.z, v.w)));
        mx = fmaxf(mx, fmaxf(fmaxf(v.x, v.y), fmaxf(v.z, v.w)));
    }

    // intra-wave reduction (wave32)
#pragma unroll
    for (int m = 16; m >= 1; m >>= 1) {
        s  += __shfl_xor(s,  m, 32);
        ss += __shfl_xor(ss, m, 32);
        mn  = fminf(mn, __shfl_xor(mn, m, 32));
        mx  = fmaxf(mx, __shfl_xor(mx, m, 32));
    }

    __shared__ float red[8][4];
    const int lane = threadIdx.x & 31;
    const int wv   = threadIdx.x >> 5;
    if (lane == 0) { red[wv][0] = s; red[wv][1] = ss; red[wv][2] = mn; red[wv][3] = mx; }
    __syncthreads();

    if (wv == 0) {
        float s2  = (lane < 8) ? red[lane][0] :  0.f;
        float ss2 = (lane < 8) ? red[lane][1] :  0.f;
        float mn2 = (lane < 8) ? red[lane][2] :  __builtin_inff();
        float mx2 = (lane < 8) ? red[lane][3] : -__builtin_inff();
#pragma unroll
        for (int m = 4; m >= 1; m >>= 1) {
            s2  += __shfl_xor(s2,  m, 32);
            ss2 += __shfl_xor(ss2, m, 32);
            mn2  = fminf(mn2, __shfl_xor(mn2, m, 32));
            mx2  = fmaxf(mx2, __shfl_xor(mx2, m, 32));
        }
        if (lane == 0) {
            v4f o; o.x = s2; o.y = ss2; o.z = mn2; o.w = mx2;
            ((v4f*)partials)[(long long)row * CHUNKS + chunk] = o;
        }
    }
}

// ---------------------------------------------------------------------------
// Pass 2: one wave per row. Cross-lane summation done with
// V_WMMA_F32_16X16X4_F32 against an all-ones B operand:
//   A layout (16x4 f32): lane l<16  -> A[l][0]=VGPR0, A[l][1]=VGPR1
//                        lane l>=16 -> A[l-16][2]=VGPR0, A[l-16][3]=VGPR1
//   With B==1:  D[m][n] = s_m + s_{m+16}  (our VGPR1 is 0)
//   Wave total = sum_{r=0..7} D_vgpr[r]  (lanes 0-15 give m=0..7,
//   lanes 16-31 give m=8..15)  then one shfl_xor(16) add.
// Emits params[row*32 + {0..14:new_q, 16..29:edges, 30:qmin, 31:qmax}]
// ---------------------------------------------------------------------------
__global__ __launch_bounds__(32)
void finalize_kernel(const float* __restrict__ partials, float* __restrict__ params,
                     float fN) {
    const int row  = blockIdx.x;
    const int lane = threadIdx.x;          // 32 threads, one wave, EXEC all ones
    const v4f* __restrict__ pr = (const v4f*)partials + (long long)row * CHUNKS;

    float s = 0.f, ss = 0.f;
    float mn = __builtin_inff(), mx = -__builtin_inff();
#pragma unroll
    for (int p = lane; p < CHUNKS; p += 32) {
        v4f v = pr[p];
        s += v.x; ss += v.y;
        mn = fminf(mn, v.z); mx = fmaxf(mx, v.w);
    }

    // WMMA reduction of s and ss across the wave
    v2f a1; a1.x = s;  a1.y = 0.f;
    v2f a2; a2.x = ss; a2.y = 0.f;
    v2f bo; bo.x = 1.f; bo.y = 1.f;
    v8f cz = {};
    v8f d1 = __builtin_amdgcn_wmma_f32_16x16x4_f32(false, a1, false, bo, (short)0, cz, false, false);
    v8f d2 = __builtin_amdgcn_wmma_f32_16x16x4_f32(false, a2, false, bo, (short)0, cz, false, false);
    float ts  = ((d1[0] + d1[1]) + (d1[2] + d1[3])) + ((d1[4] + d1[5]) + (d1[6] + d1[7]));
    float tss = ((d2[0] + d2[1]) + (d2[2] + d2[3])) + ((d2[4] + d2[5]) + (d2[6] + d2[7]));
    ts  += __shfl_xor(ts,  16, 32);
    tss += __shfl_xor(tss, 16, 32);

    // min/max via shfl tree
#pragma unroll
    for (int m = 16; m >= 1; m >>= 1) {
        mn = fminf(mn, __shfl_xor(mn, m, 32));
        mx = fmaxf(mx, __shfl_xor(mx, m, 32));
    }

    const float mean = ts / fN;
    float var = tss / fN - mean * mean;
    var = fmaxf(var, 0.f);
    const float sd = fmaxf(sqrtf(var), 1e-10f);   // jnp.clip(std, 1e-10)

    // lane j (<15) owns new_q[j]
    float nq = fmaf(Q4c[lane & 15], sd, mean);
    float nq_next = __shfl_down(nq, 1, 32);

    // q_min = min over j of (nq_j >= r_min ? nq_j : +inf)
    // q_max = max over j of (nq_j <= r_max ? nq_j : -inf)
    float vmin = (lane < 15 && nq >= mn) ? nq :  __builtin_inff();
    float vmax = (lane < 15 && nq <= mx) ? nq : -__builtin_inff();
#pragma unroll
    for (int m = 16; m >= 1; m >>= 1) {
        vmin = fminf(vmin, __shfl_xor(vmin, m, 32));
        vmax = fmaxf(vmax, __shfl_xor(vmax, m, 32));
    }

    float* __restrict__ pp = params + (long long)row * 32;
    if (lane < 15) pp[lane] = nq;
    if (lane < 14) pp[16 + lane] = 0.5f * (nq + nq_next);
    if (lane == 0) { pp[30] = vmin; pp[31] = vmax; }
}

// ---------------------------------------------------------------------------
// Pass 3: streaming quantize. searchsorted(side='left') == count(edge < x),
// realized as a 14-step select cascade (edges ascending), then clamp.
// ---------------------------------------------------------------------------
struct RowParams { float nq[15]; float e[14]; float qmin, qmax; };

__device__ __forceinline__ float quant1(const RowParams& P, float x) {
    float q = P.nq[0];
#pragma unroll
    for (int j = 0; j < 14; ++j)
        q = (x > P.e[j]) ? P.nq[j + 1] : q;     // strict: edge < x
    return fminf(fmaxf(q, P.qmin), P.qmax);     // == jnp.clip order
}

__global__ __launch_bounds__(TPB)
void quant_kernel(const float* __restrict__ x, const float* __restrict__ params,
                  float* __restrict__ out, int n4, int rowStride4) {
    const int row = blockIdx.y;
    const float* __restrict__ pb = params + (long long)row * 32;   // uniform -> SGPRs

    RowParams P;
#pragma unroll
    for (int j = 0; j < 15; ++j) P.nq[j] = pb[j];
#pragma unroll
    for (int j = 0; j < 14; ++j) P.e[j] = pb[16 + j];
    P.qmin = pb[30]; P.qmax = pb[31];

    const v4f* __restrict__ xr = (const v4f*)x   + (long long)row * n4;
    v4f* __restrict__       od = (v4f*)out       + (long long)row * n4;

    for (int i = blockIdx.x * TPB + (int)threadIdx.x; i < n4; i += rowStride4) {
        v4f v = __builtin_nontemporal_load(xr + i);   // last-use read
        v4f q;
        q.x = quant1(P, v.x);
        q.y = quant1(P, v.y);
        q.z = quant1(P, v.z);
        q.w = quant1(P, v.w);
        __builtin_nontemporal_store(q, od + i);       // streaming write, keep L2 clean
    }
}

// ---------------------------------------------------------------------------
extern "C" void kernel_launch(void* const* d_in, const int* in_sizes, int n_in,
                              void* d_out, int out_size, void* d_ws, size_t ws_size,
                              hipStream_t stream) {
    const float* x = (const float*)d_in[0];
    float* out = (float*)d_out;

    const long long total = (long long)in_sizes[0];   // 32*16*3*224*224
    const int N  = (int)(total / BROWS);              // 2,408,448
    const int n4 = N / 4;                             // 602,112 (divisible)

    float* partials = (float*)d_ws;                         // 32*128*4 floats (64 KB)
    float* params   = partials + (size_t)BROWS * CHUNKS * 4; // 32*32 floats (4 KB)

    dim3 gA(CHUNKS, BROWS);
    stats_kernel<<<gA, TPB, 0, stream>>>(x, partials, n4);

    finalize_kernel<<<BROWS, 32, 0, stream>>>(partials, params, (float)N);

    const int ipt = 16;                                   // float4s per thread
    const int threadsPerRow = (n4 + ipt - 1) / ipt;
    const int bpr = (threadsPerRow + TPB - 1) / TPB;      // 147 blocks/row
    dim3 gC(bpr, BROWS);
    quant_kernel<<<gC, TPB, 0, stream>>>(x, params, out, n4, bpr * TPB);
}